// AdaptiveConv_42099269435607
// MI455X (gfx1250) — compile-verified
//
#include <hip/hip_runtime.h>
#include <hip/hip_bf16.h>

// Problem constants (from reference)
#define BATCH 4
#define NPTS  8192
#define CIN   128
#define COUT  256
#define KNN   32
#define NW    4
#define BN_EPS 1e-5f

typedef __attribute__((ext_vector_type(2))) float v2f;
typedef __attribute__((ext_vector_type(8))) float v8f;
typedef __attribute__((ext_vector_type(4))) unsigned int u32x4;
typedef __attribute__((ext_vector_type(8))) int i32x8;
typedef __attribute__((ext_vector_type(4))) int i32x4;

// ---------------------------------------------------------------------------
// Kernel 1: kNN (top-32 smallest squared distances per query point).
// One thread per query; candidates tiled through LDS; per-thread 32-entry
// max-heap in LDS holding (dist|idx) packed as an orderable u64 key.
// ---------------------------------------------------------------------------
__global__ __launch_bounds__(128) void knn_kernel(const float* __restrict__ xyz,
                                                  int* __restrict__ idx_out) {
    __shared__ float4 cand[128];
    __shared__ unsigned long long heap[KNN * 128];  // heap[e*128 + tid]

    const int tid = threadIdx.x;
    const int q   = blockIdx.x * 128 + tid;         // global query id
    const int b   = q / NPTS;
    const int n   = q - b * NPTS;
    const float* xb = xyz + (size_t)b * NPTS * 3;

    const float qx = xb[n * 3 + 0];
    const float qy = xb[n * 3 + 1];
    const float qz = xb[n * 3 + 2];
    const float qs = qx * qx + qy * qy + qz * qz;

    for (int e = 0; e < KNN; ++e)
        heap[e * 128 + tid] = 0xFFFFFFFFFFFFFFFFull;

    for (int j0 = 0; j0 < NPTS; j0 += 128) {
        __syncthreads();
        {
            const int j = j0 + tid;
            const float cx = xb[j * 3 + 0];
            const float cy = xb[j * 3 + 1];
            const float cz = xb[j * 3 + 2];
            cand[tid] = make_float4(cx, cy, cz, cx * cx + cy * cy + cz * cz);
        }
        __syncthreads();
        for (int jj = 0; jj < 128; ++jj) {
            const float4 c = cand[jj];
            const float d = qs + c.w - 2.0f * (qx * c.x + qy * c.y + qz * c.z);
            // monotonic float->uint transform (handles tiny negatives)
            unsigned int u = __float_as_uint(d);
            u = (u & 0x80000000u) ? ~u : (u | 0x80000000u);
            const unsigned long long key =
                ((unsigned long long)u << 32) | (unsigned int)(j0 + jj);
            if (key < heap[tid]) {  // smaller than current max -> insert
                int i = 0;
                for (;;) {
                    const int l = 2 * i + 1;
                    if (l >= KNN) break;
                    int ci = l;
                    unsigned long long hc = heap[l * 128 + tid];
                    const int r = l + 1;
                    if (r < KNN) {
                        const unsigned long long hr = heap[r * 128 + tid];
                        if (hr > hc) { ci = r; hc = hr; }
                    }
                    if (hc > key) { heap[i * 128 + tid] = hc; i = ci; }
                    else break;
                }
                heap[i * 128 + tid] = key;
            }
        }
    }
    int* op = idx_out + (size_t)q * KNN;
    for (int e = 0; e < KNN; ++e)
        op[e] = (int)(heap[e * 128 + tid] & 0xFFFFFFFFu);
}

// ---------------------------------------------------------------------------
// Kernel 2: neighbor-mean features (coalesced across channels), geo-mean,
// tiny MLP (3->64 ReLU ->4) + softmax -> adaptive weights.
// One block (128 threads == CIN) per query point.
// ---------------------------------------------------------------------------
__global__ __launch_bounds__(128) void gather_kernel(
    const float* __restrict__ xyz, const float* __restrict__ points,
    const float* __restrict__ W1, const float* __restrict__ b1,
    const float* __restrict__ W2, const float* __restrict__ b2,
    const int* __restrict__ idx, float* __restrict__ feats,
    float* __restrict__ aw) {
    __shared__ int   sidx[KNN];
    __shared__ float sg[KNN * 3];
    __shared__ float sgm[3];
    __shared__ float sh[64];

    const int q   = blockIdx.x;
    const int b   = q / NPTS;
    const int n   = q - b * NPTS;
    const int tid = threadIdx.x;

    if (tid < KNN) sidx[tid] = idx[(size_t)q * KNN + tid];
    __syncthreads();

    // feats: mean over K neighbors, channel = tid (coalesced per neighbor row)
    const float* pb = points + (size_t)b * NPTS * CIN;
    float acc = 0.0f;
    for (int k = 0; k < KNN; ++k)
        acc += pb[(size_t)sidx[k] * CIN + tid];
    feats[(size_t)q * CIN + tid] = acc * (1.0f / KNN);

    // gather neighbor xyz
    const float* xb = xyz + (size_t)b * NPTS * 3;
    if (tid < KNN) {
        const int j = sidx[tid];
        sg[tid * 3 + 0] = xb[j * 3 + 0];
        sg[tid * 3 + 1] = xb[j * 3 + 1];
        sg[tid * 3 + 2] = xb[j * 3 + 2];
    }
    __syncthreads();
    if (tid < 3) {
        float s = 0.0f;
        for (int k = 0; k < KNN; ++k) s += sg[k * 3 + tid];
        sgm[tid] = s * (1.0f / KNN) - xb[n * 3 + tid];
    }
    __syncthreads();
    if (tid < 64) {
        float h = b1[tid];
        for (int d = 0; d < 3; ++d) h += sgm[d] * W1[d * 64 + tid];
        sh[tid] = fmaxf(h, 0.0f);
    }
    __syncthreads();
    if (tid == 0) {
        float lo[NW];
        for (int w = 0; w < NW; ++w) {
            float s = b2[w];
            for (int j = 0; j < 64; ++j) s += sh[j] * W2[j * NW + w];
            lo[w] = s;
        }
        float m = lo[0];
        for (int w = 1; w < NW; ++w) m = fmaxf(m, lo[w]);
        float e[NW], se = 0.0f;
        for (int w = 0; w < NW; ++w) { e[w] = __expf(lo[w] - m); se += e[w]; }
        const float inv = 1.0f / se;
        for (int w = 0; w < NW; ++w) aw[(size_t)q * NW + w] = e[w] * inv;
    }
}

// ---------------------------------------------------------------------------
// Kernel 3: folded adaptive-conv GEMM with V_WMMA_F32_16X16X4_F32.
//   C(32768x256) = A'(32768x512) @ Kflat(512x256)
//   A'[n, w*128+c] = aw[n,w] * feats[n,c]; Kflat = kernels input verbatim.
// Block: 256 threads (8 wave32), M-tile 64, full N=256.
// B chunks (16x256 f32) are staged into LDS by the Tensor Data Mover:
// one descriptor per chunk, issued by wave 0, with TDM hardware padding
// (8 dwords after every 256 dwords) reproducing the BSTRIDE=264 layout.
// ---------------------------------------------------------------------------
#define MT 64
#define ASTRIDE 132   // 64-bank conflict-free A fragment loads
#define BSTRIDE 264   // disjoint bank halves for B fragment loads
#define KCHUNK 16

__global__ __launch_bounds__(256) void gemm_kernel(
    const float* __restrict__ feats, const float* __restrict__ aw,
    const float* __restrict__ kern, float* __restrict__ out) {
    __shared__ float lA[MT * ASTRIDE];       // ~33.8 KB
    __shared__ float lB[KCHUNK * BSTRIDE];   // ~16.9 KB
    __shared__ float lAw[MT * NW];

    const int tid  = threadIdx.x;
    const int wv   = tid >> 5;
    const int lane = tid & 31;
    const int half = lane >> 4;
    const int l15  = lane & 15;
    const int m0   = blockIdx.x * MT;

    // stage A: 64 rows x 128 cols (float4 loads, padded LDS scatter)
    for (int f = tid; f < MT * CIN / 4; f += 256) {
        const int row = f >> 5;             // 32 float4 per row
        const int c4  = (f & 31) << 2;
        const float4 v = *(const float4*)(feats + (size_t)(m0 + row) * CIN + c4);
        float* dst = &lA[row * ASTRIDE + c4];
        dst[0] = v.x; dst[1] = v.y; dst[2] = v.z; dst[3] = v.w;
    }
    if (tid < MT) {
        const float4 v = *(const float4*)(aw + (size_t)(m0 + tid) * NW);
        lAw[tid * 4 + 0] = v.x; lAw[tid * 4 + 1] = v.y;
        lAw[tid * 4 + 2] = v.z; lAw[tid * 4 + 3] = v.w;
    }

    v8f acc[4][2] = {};

    for (int kb0 = 0; kb0 < NW * CIN; kb0 += KCHUNK) {
        __syncthreads();   // previous chunk fully consumed before overwrite

#if __has_builtin(__builtin_amdgcn_tensor_load_to_lds)
        if (wv == 0) {
            // --- TDM descriptor: 2-D tile, 16 rows x 256 f32, row stride 256,
            //     LDS padding 8 dwords after every 256 dwords (-> BSTRIDE=264).
            const unsigned long long gaddr =
                (unsigned long long)(uintptr_t)(kern + (size_t)kb0 * COUT);
            const unsigned int lds_off = (unsigned int)(uintptr_t)&lB[0];

            u32x4 g0;
            g0[0] = 1u;                                   // count=1, user mode
            g0[1] = lds_off;                              // lds_addr (bytes)
            g0[2] = (unsigned int)(gaddr & 0xFFFFFFFFull);            // global_addr lo
            g0[3] = (unsigned int)((gaddr >> 32) & 0x01FFFFFFull)     // global_addr hi
                  | (2u << 30);                                       // type=2 ("image")

            i32x8 g1;
            g1[0] = (2 << 16)      // data_size = 4 bytes
                  | (1 << 20)      // pad_enable
                  | (7 << 22)      // pad_interval: 256 dwords
                  | (7 << 25);     // pad_amount:   8 dwords
            g1[1] = (int)(256u << 16);   // tensor_dim0[15:0] << 16   (width 256)
            g1[2] = (int)(512u << 16);   // tensor_dim0 hi=0 | tensor_dim1 lo (512)
            g1[3] = (int)(256u << 16);   // tensor_dim1 hi=0 | tile_dim0 = 256
            g1[4] = KCHUNK;              // tile_dim1 = 16 rows, tile_dim2 = 0
            g1[5] = 256;                 // tensor_dim0_stride lo32 = 256
            g1[6] = 0;                   // stride0 hi | tensor_dim1_stride lo
            g1[7] = 0;

            i32x4 gz = {0, 0, 0, 0};     // groups 2/3 unused (2-D tensor)

#if defined(__clang_major__) && (__clang_major__ >= 23)
            i32x8 gz8 = {0, 0, 0, 0, 0, 0, 0, 0};
            __builtin_amdgcn_tensor_load_to_lds(g0, g1, gz, gz, gz8, 0);
#else
            __builtin_amdgcn_tensor_load_to_lds(g0, g1, gz, gz, 0);
#endif
            __builtin_amdgcn_s_wait_tensorcnt(0);
        }
#else
        // Fallback: cooperative vector loads into the padded layout.
        for (int f = tid; f < KCHUNK * COUT / 4; f += 256) {
            const int row = f >> 6;         // 64 float4 per row
            const int c4  = (f & 63) << 2;
            const float4 v = *(const float4*)(kern + (size_t)(kb0 + row) * COUT + c4);
            float* dst = &lB[row * BSTRIDE + c4];
            dst[0] = v.x; dst[1] = v.y; dst[2] = v.z; dst[3] = v.w;
        }
#endif
        __syncthreads();   // B chunk visible to all waves

        const int widx = kb0 >> 7;          // which adaptive kernel (constant/chunk)
        float awr[4];
        for (int m = 0; m < 4; ++m) awr[m] = lAw[(m * 16 + l15) * 4 + widx];

        for (int ks = 0; ks < KCHUNK / 4; ++ks) {
            const int c0 = (kb0 & (CIN - 1)) + ks * 4 + half * 2;  // feats column
            const int kr = ks * 4 + half * 2;                      // B chunk row
            v2f a[4];
            for (int m = 0; m < 4; ++m) {
                const v2f t = *(const v2f*)&lA[(m * 16 + l15) * ASTRIDE + c0];
                a[m] = t * awr[m];
            }
            v2f bf[2];
            for (int ni = 0; ni < 2; ++ni) {
                const int N = (wv * 2 + ni) * 16 + l15;
                bf[ni].x = lB[kr * BSTRIDE + N];
                bf[ni].y = lB[(kr + 1) * BSTRIDE + N];
            }
            for (int m = 0; m < 4; ++m)
                for (int ni = 0; ni < 2; ++ni)
                    acc[m][ni] = __builtin_amdgcn_wmma_f32_16x16x4_f32(
                        false, a[m], false, bf[ni], (short)0, acc[m][ni],
                        false, false);
        }
    }

    // epilogue: C/D layout — VGPR r: lanes0-15 M=r, lanes16-31 M=r+8
    for (int m = 0; m < 4; ++m) {
        for (int ni = 0; ni < 2; ++ni) {
            const int Nc = (wv * 2 + ni) * 16 + l15;
            for (int r = 0; r < 8; ++r) {
                const int M = m0 + m * 16 + r + half * 8;
                out[(size_t)M * COUT + Nc] = acc[m][ni][r];
            }
        }
    }
}

// ---------------------------------------------------------------------------
// BatchNorm (training-mode batch stats), two-pass.
// ---------------------------------------------------------------------------
__global__ void zero_stats(float* __restrict__ stats) {
    stats[threadIdx.x] = 0.0f;   // 512 threads: sum[256] + sumsq[256]
}

__global__ __launch_bounds__(256) void bn_stats(const float* __restrict__ out,
                                                float* __restrict__ stats) {
    const int c  = threadIdx.x;          // channel
    const int r0 = blockIdx.x * 128;     // 256 blocks x 128 rows = 32768
    float s = 0.0f, s2 = 0.0f;
    for (int r = 0; r < 128; ++r) {
        const float v = out[(size_t)(r0 + r) * COUT + c];
        s += v; s2 += v * v;
    }
    atomicAdd(&stats[c], s);
    atomicAdd(&stats[COUT + c], s2);
}

__global__ __launch_bounds__(256) void bn_apply(float* __restrict__ out,
                                                const float* __restrict__ stats,
                                                const float* __restrict__ gamma,
                                                const float* __restrict__ beta) {
    const size_t i = (size_t)blockIdx.x * 256 + threadIdx.x;
    const int c = (int)(i & (COUT - 1));
    const float inv_n = 1.0f / (float)(BATCH * NPTS);
    const float mu  = stats[c] * inv_n;
    const float var = stats[COUT + c] * inv_n - mu * mu;
    const float v = out[i];
    out[i] = (v - mu) * rsqrtf(var + BN_EPS) * gamma[c] + beta[c];
}

// ---------------------------------------------------------------------------
extern "C" void kernel_launch(void* const* d_in, const int* in_sizes, int n_in,
                              void* d_out, int out_size, void* d_ws, size_t ws_size,
                              hipStream_t stream) {
    const float* xyz    = (const float*)d_in[0];
    const float* points = (const float*)d_in[1];
    const float* W1     = (const float*)d_in[2];
    const float* b1     = (const float*)d_in[3];
    const float* W2     = (const float*)d_in[4];
    const float* b2     = (const float*)d_in[5];
    const float* kern   = (const float*)d_in[6];   // (NW, CIN, COUT) == (512, 256) row-major
    const float* gamma  = (const float*)d_in[7];
    const float* beta   = (const float*)d_in[8];
    float* out = (float*)d_out;

    char* ws = (char*)d_ws;
    int*   idx   = (int*)ws;                             // 32768*32 ints  = 4 MB
    float* feats = (float*)(ws + (size_t)(4u << 20));    // 32768*128 f32  = 16 MB
    float* aw    = (float*)(ws + (size_t)(20u << 20));   // 32768*4 f32    = 0.5 MB
    float* stats = (float*)(ws + (size_t)(21u << 20));   // 512 f32

    const int Q = BATCH * NPTS;  // 32768 query points

    knn_kernel<<<Q / 128, 128, 0, stream>>>(xyz, idx);
    gather_kernel<<<Q, 128, 0, stream>>>(xyz, points, W1, b1, W2, b2,
                                         idx, feats, aw);
    gemm_kernel<<<Q / MT, 256, 0, stream>>>(feats, aw, kern, out);
    zero_stats<<<1, 512, 0, stream>>>(stats);
    bn_stats<<<Q / 128, 256, 0, stream>>>(out, stats);
    bn_apply<<<(Q * COUT) / 256, 256, 0, stream>>>(out, stats, gamma, beta);
}